// GCN_65008624993013
// MI455X (gfx1250) — compile-verified
//
#include <hip/hip_runtime.h>

#define N_NODES 200000
#define N_EDGES 6400000
#define F_IN    165
#define F_HID   16
#define F_OUT   2

typedef __attribute__((ext_vector_type(2))) float v2f;
typedef __attribute__((ext_vector_type(8))) float v8f;

// ---------------------------------------------------------------------------
// Degree: deg[i] = 1 (self loop) + #edges with dst == i ; then dinv = rsqrt.
// ---------------------------------------------------------------------------
__global__ void k_deg_init(float* __restrict__ deg) {
    int i = blockIdx.x * blockDim.x + threadIdx.x;
    if (i < N_NODES) deg[i] = 1.0f;
}

__global__ void k_deg_count(const long long* __restrict__ dst, float* __restrict__ deg) {
    int e = blockIdx.x * blockDim.x + threadIdx.x;
    if (e < N_EDGES) {
        int d = (int)dst[e];
        atomicAdd(&deg[d], 1.0f);
    }
}

__global__ void k_dinv(float* __restrict__ deg) {
    int i = blockIdx.x * blockDim.x + threadIdx.x;
    if (i < N_NODES) deg[i] = rsqrtf(deg[i]);   // deg >= 1 always
}

// ---------------------------------------------------------------------------
// h1 = x @ W1   (200000x165 @ 165x16) via V_WMMA_F32_16X16X4_F32.
// One wave per 16x16 output tile; 12500 tiles total. blockDim=128 (4 waves),
// grid=3125 -> exactly 12500 waves, EXEC all-ones for every WMMA.
// K split: 41 full K=4 steps (no guards) + one branch-free tail for K=164.
// Fragment layouts per CDNA5 ISA 7.12.2 (32-bit A 16x4 / B 4x16 / D 16x16).
// ---------------------------------------------------------------------------
__global__ void k_gemm1(const float* __restrict__ x, const float* __restrict__ W1,
                        float* __restrict__ h1) {
    int wave = blockIdx.x * 4 + (threadIdx.x >> 5);   // 0 .. 12499
    int lane = threadIdx.x & 31;
    int half = lane >> 4;           // 0: lanes 0-15, 1: lanes 16-31
    int l15  = lane & 15;

    int rowBase = wave * 16;
    const float* xr = x + (long)(rowBase + l15) * F_IN + half * 2; // A row, this half's K pair
    const float* bp = W1 + l15 + half * 2 * F_HID;                 // B col, this half's K pair

    v8f acc = {};
#pragma unroll 2
    for (int k = 0; k < F_IN - 1; k += 4) {          // k = 0,4,...,160 : all full
        v2f a, b;
        a.x = xr[k];                                  // A[M=l15, K=k+2*half]
        a.y = xr[k + 1];                              // A[M=l15, K=k+2*half+1]
        b.x = bp[(long)k * F_HID];                    // B[K=k+2*half,   N=l15]
        b.y = bp[(long)(k + 1) * F_HID];              // B[K=k+2*half+1, N=l15]
        acc = __builtin_amdgcn_wmma_f32_16x16x4_f32(
                  false, a, false, b, (short)0, acc, false, false);
    }

    // Tail: K = 164 only (valid in lower lane-half's first slot). Unconditional
    // in-bounds loads + branch-free select; no EXEC divergence around WMMA.
    {
        float ta = x[(long)(rowBase + l15) * F_IN + (F_IN - 1)];
        float tb = W1[(F_IN - 1) * F_HID + l15];
        v2f a, b;
        a.x = half ? 0.0f : ta;
        a.y = 0.0f;
        b.x = half ? 0.0f : tb;
        b.y = 0.0f;
        acc = __builtin_amdgcn_wmma_f32_16x16x4_f32(
                  false, a, false, b, (short)0, acc, false, false);
    }

    // D layout: VGPR r -> row = rowBase + r + 8*half, col = l15
    int rRow = rowBase + half * 8;
#pragma unroll
    for (int r = 0; r < 8; ++r) {
        h1[(long)(rRow + r) * F_HID + l15] = acc[r];
    }
}

// ---------------------------------------------------------------------------
// agg1[i,:] = h1[i,:] * dinv[i]^2        (self-loop term, also zero-inits)
// ---------------------------------------------------------------------------
__global__ void k_agg1_init(const float* __restrict__ h1, const float* __restrict__ dinv,
                            float* __restrict__ agg1) {
    int t = blockIdx.x * blockDim.x + threadIdx.x;   // one thread per (node, feature)
    if (t < N_NODES * F_HID) {
        int i = t >> 4;
        float di = dinv[i];
        agg1[t] = h1[t] * (di * di);
    }
}

// ---------------------------------------------------------------------------
// Edge scatter, layer 1: 4 threads per edge, 4 features each.
// agg1[dst,:] += h1[src,:] * dinv[src]*dinv[dst]
// ---------------------------------------------------------------------------
__global__ void k_scatter1(const long long* __restrict__ ei, const float* __restrict__ h1,
                           const float* __restrict__ dinv, float* __restrict__ agg1) {
    long long t = (long long)blockIdx.x * blockDim.x + threadIdx.x;
    if (t >= (long long)N_EDGES * 4) return;
    int e = (int)(t >> 2);
    int q = (int)(t & 3);
    int s = (int)ei[e];
    int d = (int)ei[N_EDGES + e];
    float nrm = dinv[s] * dinv[d];
    const float4 hv = *(const float4*)(h1 + (long)s * F_HID + q * 4);
    float* ap = agg1 + (long)d * F_HID + q * 4;
    atomicAdd(ap + 0, hv.x * nrm);
    atomicAdd(ap + 1, hv.y * nrm);
    atomicAdd(ap + 2, hv.z * nrm);
    atomicAdd(ap + 3, hv.w * nrm);
}

// ---------------------------------------------------------------------------
// h2[i,:] = relu(agg1[i,:] + b1) @ W2      (16 -> 2, scalar FMA per node)
// ---------------------------------------------------------------------------
__global__ void k_layer2(const float* __restrict__ agg1, const float* __restrict__ b1,
                         const float* __restrict__ W2, float* __restrict__ h2) {
    int i = blockIdx.x * blockDim.x + threadIdx.x;
    if (i >= N_NODES) return;
    const float* ar = agg1 + (long)i * F_HID;
    float o0 = 0.0f, o1 = 0.0f;
#pragma unroll
    for (int f = 0; f < F_HID; ++f) {
        float v = ar[f] + b1[f];
        v = fmaxf(v, 0.0f);
        o0 = fmaf(v, W2[f * F_OUT + 0], o0);
        o1 = fmaf(v, W2[f * F_OUT + 1], o1);
    }
    h2[i * F_OUT + 0] = o0;
    h2[i * F_OUT + 1] = o1;
}

// ---------------------------------------------------------------------------
// out[i,:] = h2[i,:] * dinv[i]^2 + b2      (self-loop term + bias, inits out)
// ---------------------------------------------------------------------------
__global__ void k_out_init(const float* __restrict__ h2, const float* __restrict__ dinv,
                           const float* __restrict__ b2, float* __restrict__ out) {
    int i = blockIdx.x * blockDim.x + threadIdx.x;
    if (i >= N_NODES) return;
    float di = dinv[i];
    float d2 = di * di;
    out[i * F_OUT + 0] = h2[i * F_OUT + 0] * d2 + b2[0];
    out[i * F_OUT + 1] = h2[i * F_OUT + 1] * d2 + b2[1];
}

// ---------------------------------------------------------------------------
// Edge scatter, layer 2: 1 thread per edge, 2 features.
// ---------------------------------------------------------------------------
__global__ void k_scatter2(const long long* __restrict__ ei, const float* __restrict__ h2,
                           const float* __restrict__ dinv, float* __restrict__ out) {
    int e = blockIdx.x * blockDim.x + threadIdx.x;
    if (e >= N_EDGES) return;
    int s = (int)ei[e];
    int d = (int)ei[N_EDGES + e];
    float nrm = dinv[s] * dinv[d];
    atomicAdd(out + (long)d * F_OUT + 0, h2[s * F_OUT + 0] * nrm);
    atomicAdd(out + (long)d * F_OUT + 1, h2[s * F_OUT + 1] * nrm);
}

// ---------------------------------------------------------------------------
extern "C" void kernel_launch(void* const* d_in, const int* in_sizes, int n_in,
                              void* d_out, int out_size, void* d_ws, size_t ws_size,
                              hipStream_t stream) {
    const float*     x  = (const float*)d_in[0];                 // [200000,165]
    const long long* ei = (const long long*)d_in[1];             // [2,6400000] int64
    const float*     W1 = (const float*)d_in[2];                 // [165,16]
    const float*     b1 = (const float*)d_in[3];                 // [16]
    const float*     W2 = (const float*)d_in[4];                 // [16,2]
    const float*     b2 = (const float*)d_in[5];                 // [2]
    float*           out = (float*)d_out;                        // [200000,2]

    // Workspace layout (floats): dinv | h1 | agg1 | h2  = 7,000,000 floats (28 MB)
    float* dinv = (float*)d_ws;
    float* h1   = dinv + N_NODES;            // 200000*16
    float* agg1 = h1   + N_NODES * F_HID;    // 200000*16
    float* h2   = agg1 + N_NODES * F_HID;    // 200000*2

    const int B = 256;

    // Degree / dinv (shared by both layers; edge_index identical)
    k_deg_init <<<(N_NODES + B - 1) / B, B, 0, stream>>>(dinv);
    k_deg_count<<<(N_EDGES + B - 1) / B, B, 0, stream>>>(ei + N_EDGES, dinv);
    k_dinv     <<<(N_NODES + B - 1) / B, B, 0, stream>>>(dinv);

    // Layer 1 dense: h1 = x @ W1 (WMMA, one wave per 16x16 tile)
    k_gemm1<<<N_NODES / 16 / 4, 128, 0, stream>>>(x, W1, h1);

    // Layer 1 aggregation
    k_agg1_init<<<(N_NODES * F_HID + B - 1) / B, B, 0, stream>>>(h1, dinv, agg1);
    long long s1_threads = (long long)N_EDGES * 4;
    k_scatter1<<<(unsigned)((s1_threads + B - 1) / B), B, 0, stream>>>(ei, h1, dinv, agg1);

    // Layer 2 dense: h2 = relu(agg1 + b1) @ W2
    k_layer2<<<(N_NODES + B - 1) / B, B, 0, stream>>>(agg1, b1, W2, h2);

    // Layer 2 aggregation + bias -> out
    k_out_init<<<(N_NODES + B - 1) / B, B, 0, stream>>>(h2, dinv, b2, out);
    k_scatter2<<<(N_EDGES + B - 1) / B, B, 0, stream>>>(ei, h2, dinv, out);
}